// Sampler_21603685499602
// MI455X (gfx1250) — compile-verified
//
#include <hip/hip_runtime.h>

typedef __attribute__((ext_vector_type(2))) float v2f;
typedef __attribute__((ext_vector_type(8))) float v8f;

#define NPOINT 10
#define NPTS   262144
#define BIGD   1e10f

__global__ __launch_bounds__(1024) void fps_wmma_kernel(const float* __restrict__ xyz,
                                                        int* __restrict__ out) {
  const int b    = blockIdx.x;
  const int tid  = threadIdx.x;
  const int lane = tid & 31;
  const int wave = tid >> 5;          // 0..31 waves per block
  const int col  = lane & 15;         // B/D column this lane holds
  const int slot = lane & 7;          // mirrored A row slot (rows m and m+8 identical)
  const bool hi  = lane >= 16;        // K2/K3 half of A/B

  const float* base = xyz + (size_t)b * NPTS * 3;

  __shared__ float s_wval[32];
  __shared__ int   s_widx[32];
  __shared__ float s_pt[4];           // px, py, pz, ||p||^2 broadcast

  // --- select point 0 (reference starts at index 0) ---
  if (tid == 0) {
    float px = base[0], py = base[1], pz = base[2];
    s_pt[0] = px; s_pt[1] = py; s_pt[2] = pz;
    s_pt[3] = px * px + py * py + pz * pz;
    out[b * NPOINT + 0] = 0;
  }
  __syncthreads();

  // Mirrored A matrix: row m holds selected point (m & 7), so rows 0-7 == rows 8-15.
  // Every lane's 8 D components then cover ALL selected points -> no cross-half
  // shuffle needed in the hot loop.
  //   lane (slot, lo-half): A[m][0], A[m][1] = -2px, -2py
  //   lane (slot, hi-half): A[m][2], A[m][3] = -2pz, ||p||^2
  // empty slots -> [0,0] / [0,BIGD]  (B row K3 = 1.0 => contribution BIGD)
  float a0, a1;
  {
    float px = s_pt[0], py = s_pt[1], pz = s_pt[2], pn = s_pt[3];
    if (slot == 0) {
      a0 = hi ? -2.0f * pz : -2.0f * px;
      a1 = hi ? pn         : -2.0f * py;
    } else {
      a0 = 0.0f;
      a1 = hi ? BIGD : 0.0f;
    }
  }
  __syncthreads();

  // point 8 (9th selected) doesn't fit the 8 mirrored slots; handled via VALU in
  // the final round only.
  float q0 = 0.0f, q1 = 0.0f, q2 = 0.0f;

  const int tilesPerBatch = NPTS / 16;   // 16384 tiles of 16 candidate points

  for (int k = 1; k < NPOINT; ++k) {
    float bestVal = -1.0f;
    int   bestIdx = 0;

    if (k < NPOINT - 1) {
      // rounds 1..8: all selected points live in the mirrored WMMA slots
#pragma unroll 2
      for (int t = wave; t < tilesPerBatch; t += 32) {
        const int p = t * 16 + col;
        const float* pp = base + (size_t)p * 3;
        float x = pp[0], y = pp[1], z = pp[2];
        float xn = __builtin_fmaf(x, x, __builtin_fmaf(y, y, z * z));
        // speculative prefetch 2 strides ahead; offset folds into the instruction
        __builtin_prefetch(pp + 2 * 32 * 16 * 3, 0, 1);

        v2f A;  A.x  = a0;         A.y  = a1;
        v2f Bm; Bm.x = hi ? z : x; Bm.y = hi ? 1.0f : y;
        v8f C = {};
        v8f D = __builtin_amdgcn_wmma_f32_16x16x4_f32(
            false, A, false, Bm, (short)0, C, false, false);

        // full min over all selected points, entirely in-lane
        float m = fminf(fminf(fminf(D.s0, D.s1), fminf(D.s2, D.s3)),
                        fminf(fminf(D.s4, D.s5), fminf(D.s6, D.s7)));
        float dist = xn + m;

        bool better = dist > bestVal;          // in-order scan keeps lowest idx on ties
        bestVal = better ? dist : bestVal;
        bestIdx = better ? p    : bestIdx;
      }
    } else {
      // final round: 8 WMMA slots + point 8 via a short VALU chain
#pragma unroll 2
      for (int t = wave; t < tilesPerBatch; t += 32) {
        const int p = t * 16 + col;
        const float* pp = base + (size_t)p * 3;
        float x = pp[0], y = pp[1], z = pp[2];
        float xn = __builtin_fmaf(x, x, __builtin_fmaf(y, y, z * z));
        __builtin_prefetch(pp + 2 * 32 * 16 * 3, 0, 1);

        v2f A;  A.x  = a0;         A.y  = a1;
        v2f Bm; Bm.x = hi ? z : x; Bm.y = hi ? 1.0f : y;
        v8f C = {};
        v8f D = __builtin_amdgcn_wmma_f32_16x16x4_f32(
            false, A, false, Bm, (short)0, C, false, false);

        float m = fminf(fminf(fminf(D.s0, D.s1), fminf(D.s2, D.s3)),
                        fminf(fminf(D.s4, D.s5), fminf(D.s6, D.s7)));
        float dist = xn + m;

        float dx = x - q0, dy = y - q1, dz = z - q2;
        float d8 = __builtin_fmaf(dx, dx, __builtin_fmaf(dy, dy, dz * dz));
        dist = fminf(dist, d8);

        bool better = dist > bestVal;
        bestVal = better ? dist : bestVal;
        bestIdx = better ? p    : bestIdx;
      }
    }

    // wave argmax reduction, branchless, tie-break toward lower index
#pragma unroll
    for (int off = 16; off >= 1; off >>= 1) {
      float ov = __shfl_xor(bestVal, off);
      int   oi = __shfl_xor(bestIdx, off);
      bool take = (ov > bestVal) | ((ov == bestVal) & (oi < bestIdx));
      bestVal = take ? ov : bestVal;
      bestIdx = take ? oi : bestIdx;
    }
    if (lane == 0) { s_wval[wave] = bestVal; s_widx[wave] = bestIdx; }
    __syncthreads();

    // wave 0 reduces the 32 per-wave winners
    if (wave == 0) {
      float v = s_wval[lane];
      int   i = s_widx[lane];
#pragma unroll
      for (int off = 16; off >= 1; off >>= 1) {
        float ov = __shfl_xor(v, off);
        int   oi = __shfl_xor(i, off);
        bool take = (ov > v) | ((ov == v) & (oi < i));
        v = take ? ov : v;
        i = take ? oi : i;
      }
      if (lane == 0) {
        out[b * NPOINT + k] = i;
        const float* pp = base + (size_t)i * 3;
        float px = pp[0], py = pp[1], pz = pp[2];
        s_pt[0] = px; s_pt[1] = py; s_pt[2] = pz;
        s_pt[3] = px * px + py * py + pz * pz;
      }
    }
    __syncthreads();

    if (k <= 7) {
      // install point k into mirrored slot k (rows k and k+8)
      if (slot == k) {
        float px = s_pt[0], py = s_pt[1], pz = s_pt[2], pn = s_pt[3];
        a0 = hi ? -2.0f * pz : -2.0f * px;
        a1 = hi ? pn         : -2.0f * py;
      }
    } else if (k == 8) {
      // stash point 8 for the VALU path of the final round (all threads)
      q0 = s_pt[0]; q1 = s_pt[1]; q2 = s_pt[2];
    }
    __syncthreads();
  }
}

extern "C" void kernel_launch(void* const* d_in, const int* in_sizes, int n_in,
                              void* d_out, int out_size, void* d_ws, size_t ws_size,
                              hipStream_t stream) {
  (void)n_in; (void)out_size; (void)d_ws; (void)ws_size;
  const float* xyz = (const float*)d_in[0];
  int* out = (int*)d_out;                      // reference output dtype is int32
  const int B = in_sizes[0] / (NPTS * 3);      // 64 for the given setup
  fps_wmma_kernel<<<B, 1024, 0, stream>>>(xyz, out);
}